// GINLayer_91130616087333
// MI455X (gfx1250) — compile-verified
//
#include <hip/hip_runtime.h>

typedef __attribute__((ext_vector_type(2))) float v2f;
typedef __attribute__((ext_vector_type(8))) float v8f;

#define D 128
#define WT_STRIDE 132          // pad 4: B-frag ds_load_b64 banks = 4*lane+2*hi -> conflict-free
#define TILES_PER_WG 8
#define MLP_THREADS (TILES_PER_WG * 32)

// ---------------------------------------------------------------- out = x
__global__ void gin_init_copy(const float* __restrict__ x, float* __restrict__ out, int n4) {
    int i = blockIdx.x * blockDim.x + threadIdx.x;
    if (i < n4) ((float4*)out)[i] = ((const float4*)x)[i];
}

// ------------------------------------------------- z[dst] += x[src], 1 wave/edge
__global__ void gin_scatter_add(const float* __restrict__ x, const int* __restrict__ ei,
                                float* z, int nEdges) {
    int t    = blockIdx.x * blockDim.x + threadIdx.x;
    int edge = t >> 5;
    if (edge >= nEdges) return;
    int lane = t & 31;
    int src  = ei[edge];
    int dst  = ei[edge + nEdges];
    const float4 v = ((const float4*)(x + (size_t)src * D))[lane];
    float* zp = z + (size_t)dst * D + lane * 4;
    unsafeAtomicAdd(zp + 0, v.x);
    unsafeAtomicAdd(zp + 1, v.y);
    unsafeAtomicAdd(zp + 2, v.z);
    unsafeAtomicAdd(zp + 3, v.w);
}

// ---------------------------------------------------------------- fused MLP layer
// out[16r..] = relu(in[16r..] @ W + b)  (+ (1+eps)*x if FINAL), in-place safe:
// each wave reads exactly the 16 rows it later writes; stores depend on all loads.
template <bool FINAL>
__global__ __launch_bounds__(MLP_THREADS) void gin_mlp(
    const float* in, const float* __restrict__ W, const float* __restrict__ bias,
    const float* __restrict__ xres, const float* __restrict__ epsp,
    float* out, int nTiles) {
    __shared__ float WT[D * WT_STRIDE];  // WT[n*stride + k] = W[k][n]

    // cooperative transposed weight stage (coalesced global reads)
    for (int j = threadIdx.x; j < D * D; j += MLP_THREADS) {
        int k = j >> 7, n = j & 127;
        WT[n * WT_STRIDE + k] = W[j];
    }
    __syncthreads();

    int wave = threadIdx.x >> 5;
    int lane = threadIdx.x & 31;
    int tile = blockIdx.x * TILES_PER_WG + wave;
    if (tile >= nTiles) return;        // wave-uniform: EXEC stays all-ones for WMMA

    int row0 = tile * 16;
    int m    = lane & 15;              // A: row in tile  |  B/C/D: column offset
    int hi   = lane >> 4;              // half-wave selects K pair / row+8

    const float* arow = in + (size_t)(row0 + m) * D + 2 * hi;

    v8f acc[8] = {};
    for (int kt = 0; kt < 32; ++kt) {
        // A frag 16x4 f32: lane<16 -> {K=4kt, 4kt+1}; lane>=16 -> {4kt+2, 4kt+3}
        v2f a = *(const v2f*)(arow + 4 * kt);
#pragma unroll
        for (int nt = 0; nt < 8; ++nt) {
            // B frag 4x16 f32: lane -> col 16nt+m, VGPR j -> K=4kt+2hi+j
            v2f b = *(const v2f*)(&WT[(nt * 16 + m) * WT_STRIDE + 4 * kt + 2 * hi]);
            acc[nt] = __builtin_amdgcn_wmma_f32_16x16x4_f32(
                false, a, false, b, (short)0, acc[nt], false, false);
        }
    }

    float e = 0.0f;
    if (FINAL) e = 1.0f + epsp[0];
#pragma unroll
    for (int nt = 0; nt < 8; ++nt) {
        int   n  = nt * 16 + m;
        float bn = bias[n];
#pragma unroll
        for (int r = 0; r < 8; ++r) {           // C/D: VGPR r -> row r+8*hi, lane&15 -> col
            int   row = row0 + r + 8 * hi;
            float v   = acc[nt][r] + bn;
            v = v > 0.0f ? v : 0.0f;
            if (FINAL) v += e * xres[(size_t)row * D + n];
            out[(size_t)row * D + n] = v;
        }
    }
}

// ---------------------------------------------------------------------- launch
extern "C" void kernel_launch(void* const* d_in, const int* in_sizes, int n_in,
                              void* d_out, int out_size, void* d_ws, size_t ws_size,
                              hipStream_t stream) {
    const float* x   = (const float*)d_in[0];
    const int*   ei  = (const int*)d_in[1];
    const float* W1  = (const float*)d_in[2];
    const float* b1  = (const float*)d_in[3];
    const float* W2  = (const float*)d_in[4];
    const float* b2  = (const float*)d_in[5];
    const float* eps = (const float*)d_in[6];
    float*       out = (float*)d_out;

    int N = in_sizes[0] / D;     // 100000
    int E = in_sizes[1] / 2;     // 1600000
    int nTiles = N / 16;         // 6250 (N % 16 == 0)

    // 1) z := x  (self term of GIN aggregation)
    int n4 = (N * D) / 4;
    gin_init_copy<<<dim3((n4 + 255) / 256), dim3(256), 0, stream>>>(x, out, n4);

    // 2) z[dst] += x[src] over all edges (one wave per edge)
    long long st = (long long)E * 32;
    gin_scatter_add<<<dim3((unsigned)((st + 255) / 256)), dim3(256), 0, stream>>>(x, ei, out, E);

    // 3) h1 = relu(z @ W1 + b1)   (in-place in d_out)
    int nWG = (nTiles + TILES_PER_WG - 1) / TILES_PER_WG;
    gin_mlp<false><<<dim3(nWG), dim3(MLP_THREADS), 0, stream>>>(
        out, W1, b1, nullptr, nullptr, out, nTiles);

    // 4) out = relu(h1 @ W2 + b2) + (1+eps)*x   (in-place in d_out)
    gin_mlp<true><<<dim3(nWG), dim3(MLP_THREADS), 0, stream>>>(
        out, W2, b2, x, eps, out, nTiles);
}